// multi_ligand_cob_28845000360194
// MI455X (gfx1250) — compile-verified
//
#include <hip/hip_runtime.h>
#include <math.h>

// ---------------------------------------------------------------------------
// Problem constants (mirror the JAX reference)
// ---------------------------------------------------------------------------
constexpr int   NN     = 4096;   // system size
constexpr int   NB     = 32;     // LU panel width (multiple of 4 for WMMA K)
constexpr int   NCELLS = 5;
constexpr int   DEG    = 6;

constexpr float kLRATIO = 0.57735026918962576f;   // 1/sqrt(3)
constexpr float kAREA   = 6.1836379e-09f;         // sqrt(3)/2 * (8.45e-5)^2
constexpr float kASCALE = 1.0e11f;
constexpr float kBSCALE = 1.0e23f;

typedef __attribute__((ext_vector_type(2))) float v2f;
typedef __attribute__((ext_vector_type(8))) float v8f;

// ---------------------------------------------------------------------------
// 1) Zero the dense matrix (64 MB, float4 stores)
// ---------------------------------------------------------------------------
__global__ void k_zero(float4* __restrict__ A) {
  int i = blockIdx.x * blockDim.x + threadIdx.x;
  A[i] = make_float4(0.f, 0.f, 0.f, 0.f);
}

// ---------------------------------------------------------------------------
// 2) Scatter-build A (pre-scaled by A_SCALE).  One thread per row; off-diag
//    writes first, diagonal last — identical overwrite order to the
//    reference's .set(...) calls (handles neighbour==row and duplicates).
// ---------------------------------------------------------------------------
__global__ void k_buildA(float* __restrict__ A, const float* __restrict__ th,
                         const float* __restrict__ cd, const int* __restrict__ nb) {
  int i = blockIdx.x * blockDim.x + threadIdx.x;
  if (i >= NN) return;
  const float D = th[0], lam = th[1], Kc = th[7];
  float dot = 0.f;
#pragma unroll
  for (int c = 0; c < NCELLS; ++c) dot += cd[i * NCELLS + c] * th[8 + c];
  dot *= Kc;
  const float offv = -D * kLRATIO * kASCALE;
  const float diag = (D * kLRATIO * (float)DEG + kAREA * (lam + dot)) * kASCALE;
  float* row = A + (size_t)i * NN;
#pragma unroll
  for (int j = 0; j < DEG; ++j) row[nb[i * DEG + j]] = offv;
  row[i] = diag;
}

// ---------------------------------------------------------------------------
// 3) b = AREA * (cd @ rho) * B_SCALE
// ---------------------------------------------------------------------------
__global__ void k_buildB(float* __restrict__ vec, const float* __restrict__ th,
                         const float* __restrict__ cd) {
  int i = blockIdx.x * blockDim.x + threadIdx.x;
  if (i >= NN) return;
  float dot = 0.f;
#pragma unroll
  for (int c = 0; c < NCELLS; ++c) dot += cd[i * NCELLS + c] * th[2 + c];
  vec[i] = kAREA * dot * kBSCALE;
}

// ---------------------------------------------------------------------------
// 4) Panel factor: pivot-free LU of A[K0:NN, K0:K0+NB]. Single workgroup,
//    pivot row cached in LDS, rows strided across threads.  Valid because
//    the reference matrix is diagonally dominant.
// ---------------------------------------------------------------------------
__global__ void k_panel(float* __restrict__ A, int K0) {
  __shared__ float prow[NB];
  const int tid = threadIdx.x;
  const int T   = blockDim.x;
  for (int j = 0; j < NB; ++j) {
    __syncthreads();
    if (tid < NB) prow[tid] = A[(size_t)(K0 + j) * NN + K0 + tid];
    __syncthreads();
    const float pinv = 1.0f / prow[j];
    for (int r = K0 + j + 1 + tid; r < NN; r += T) {
      float* arow = A + (size_t)r * NN + K0;
      const float l = arow[j] * pinv;
      arow[j] = l;
      for (int c = j + 1; c < NB; ++c) arow[c] -= l * prow[c];
    }
    __threadfence_block();
  }
}

// ---------------------------------------------------------------------------
// 5) TRSM: U12 = L11^{-1} * A12.  L11 (unit-lower, NB x NB) cached in LDS,
//    one column of A12 per thread, column kept in registers (fully unrolled).
// ---------------------------------------------------------------------------
__global__ void k_trsm(float* __restrict__ A, int K0) {
  __shared__ float Ls[NB][NB + 1];
  const int tid = threadIdx.x;
  for (int idx = tid; idx < NB * NB; idx += blockDim.x)
    Ls[idx / NB][idx % NB] = A[(size_t)(K0 + idx / NB) * NN + K0 + idx % NB];
  __syncthreads();
  const int ci = blockIdx.x * blockDim.x + tid;
  const int M  = NN - K0 - NB;
  if (ci >= M) return;
  const int col = K0 + NB + ci;
  float x[NB];
#pragma unroll
  for (int i = 0; i < NB; ++i) x[i] = A[(size_t)(K0 + i) * NN + col];
#pragma unroll
  for (int i = 1; i < NB; ++i) {
    float xi = x[i];
#pragma unroll
    for (int j = 0; j < i; ++j) xi -= Ls[i][j] * x[j];
    x[i] = xi;
  }
#pragma unroll
  for (int i = 0; i < NB; ++i) A[(size_t)(K0 + i) * NN + col] = x[i];
}

// ---------------------------------------------------------------------------
// 6) Trailing update (the hot GEMM): A22 -= L21 * U12 via FP32 WMMA.
//
//    Workgroup (256 thr = 8 waves) computes a 128x64 C macro-tile:
//      waves arranged 4x2, each wave owns a 32x32 tile = 2x2 accumulators.
//    L21 (128xNB) and U12 (NBx64) panels staged in LDS with coalesced
//    row-contiguous global loads; strided A-fragment reads come out of
//    padded LDS (bank-conflict-free) instead of 16 separate cachelines.
//    A-fragment negated on LDS read so D = (-L)*U + C accumulates the
//    Schur-complement subtraction directly (FP32 WMMA has no A/B NEG).
//
//    Fragment layouts per CDNA5 ISA 7.12.2 (32-bit, wave32):
//      A 16x4 : lane M = lane&15, VGPR0 -> K = half*2, VGPR1 -> K = half*2+1
//      B 4x16 : lane N = lane&15, same K striping
//      C/D    : VGPR v -> row v + half*8, col = lane&15
// ---------------------------------------------------------------------------
__global__ void k_gemm(float* __restrict__ A, int K0) {
  __shared__ float Lp[128][NB + 1];   // L21 panel   (+1 pad: conflict-free rows)
  __shared__ float Up[NB][64 + 1];    // U12 panel

  const int M       = NN - K0 - NB;        // trailing size (multiple of 32)
  const int tiles_x = (M + 63) / 64;
  const int by      = blockIdx.x / tiles_x;
  const int bx      = blockIdx.x % tiles_x;
  const int tid     = threadIdx.x;

  // ---- stage panels (coalesced, zero-padded at the ragged edge) ----
  for (int idx = tid; idx < 128 * NB; idx += 256) {
    const int r = idx >> 5, c = idx & (NB - 1);
    const int gr = by * 128 + r;
    Lp[r][c] = (gr < M) ? A[(size_t)(K0 + NB + gr) * NN + K0 + c] : 0.f;
  }
  for (int idx = tid; idx < NB * 64; idx += 256) {
    const int kk = idx >> 6, c = idx & 63;
    const int gc = bx * 64 + c;
    Up[kk][c] = (gc < M) ? A[(size_t)(K0 + kk) * NN + K0 + NB + gc] : 0.f;
  }
  __syncthreads();

  // ---- per-wave 32x32 tile ----
  const int wave = tid >> 5;
  const int lane = tid & 31;
  const int wy   = wave >> 1;              // 0..3
  const int wx   = wave & 1;               // 0..1
  const int lm   = lane & 15;
  const int half = lane >> 4;

  const int rowT = by * 128 + wy * 32;     // tile base within trailing matrix
  const int colT = bx * 64 + wx * 32;
  if (rowT < M && colT < M) {              // wave-uniform; 32 | M -> all-or-none
    const int grow = K0 + NB + rowT;
    const int gcol = K0 + NB + colT;
    __builtin_prefetch(&A[(size_t)grow * NN + gcol], 0, 0);  // global_prefetch_b8

    v8f acc[2][2];
#pragma unroll
    for (int s = 0; s < 2; ++s)
#pragma unroll
      for (int t = 0; t < 2; ++t)
#pragma unroll
        for (int v = 0; v < 8; ++v)
          acc[s][t][v] =
              A[(size_t)(grow + s * 16 + v + half * 8) * NN + gcol + t * 16 + lm];

#pragma unroll
    for (int k = 0; k < NB; k += 4) {
      v2f a[2], b[2];
#pragma unroll
      for (int s = 0; s < 2; ++s) {
        const int rowL = wy * 32 + s * 16 + lm;
        a[s].x = -Lp[rowL][k + half * 2];
        a[s].y = -Lp[rowL][k + half * 2 + 1];
      }
#pragma unroll
      for (int t = 0; t < 2; ++t) {
        const int colL = wx * 32 + t * 16 + lm;
        b[t].x = Up[k + half * 2][colL];
        b[t].y = Up[k + half * 2 + 1][colL];
      }
#pragma unroll
      for (int s = 0; s < 2; ++s)
#pragma unroll
        for (int t = 0; t < 2; ++t)
          acc[s][t] = __builtin_amdgcn_wmma_f32_16x16x4_f32(
              /*neg_a=*/false, a[s], /*neg_b=*/false, b[t],
              /*c_mod=*/(short)0, acc[s][t], /*reuse_a=*/false, /*reuse_b=*/false);
    }

#pragma unroll
    for (int s = 0; s < 2; ++s)
#pragma unroll
      for (int t = 0; t < 2; ++t)
#pragma unroll
        for (int v = 0; v < 8; ++v)
          A[(size_t)(grow + s * 16 + v + half * 8) * NN + gcol + t * 16 + lm] =
              acc[s][t][v];
  }
}

// ---------------------------------------------------------------------------
// 7) Forward substitution y = L^{-1} b (unit diag), row-dot form (coalesced).
// ---------------------------------------------------------------------------
__global__ void k_fwd(const float* __restrict__ A, float* __restrict__ vec) {
  __shared__ float red[256];
  const int tid = threadIdx.x;
  for (int i = 1; i < NN; ++i) {
    float p = 0.f;
    const float* row = A + (size_t)i * NN;
    for (int j = tid; j < i; j += 256) p += row[j] * vec[j];
    red[tid] = p;
    __syncthreads();
    for (int s = 128; s > 0; s >>= 1) {
      if (tid < s) red[tid] += red[tid + s];
      __syncthreads();
    }
    if (tid == 0) { vec[i] -= red[0]; __threadfence_block(); }
    __syncthreads();
  }
}

// ---------------------------------------------------------------------------
// 8) Backward substitution x = U^{-1} y.
// ---------------------------------------------------------------------------
__global__ void k_bwd(const float* __restrict__ A, float* __restrict__ vec) {
  __shared__ float red[256];
  const int tid = threadIdx.x;
  for (int i = NN - 1; i >= 0; --i) {
    float p = 0.f;
    const float* row = A + (size_t)i * NN;
    for (int j = i + 1 + tid; j < NN; j += 256) p += row[j] * vec[j];
    red[tid] = p;
    __syncthreads();
    for (int s = 128; s > 0; s >>= 1) {
      if (tid < s) red[tid] += red[tid + s];
      __syncthreads();
    }
    if (tid == 0) {
      vec[i] = (vec[i] - red[0]) / row[i];
      __threadfence_block();
    }
    __syncthreads();
  }
}

// ---------------------------------------------------------------------------
// 9) I = K * outer(rec, g) @ prod_term / N  ->  25 floats.
//    s[c] = sum_n g[n] * cd[n,c]*rho[c] / sum_c' cd[n,c']*rho[c']
// ---------------------------------------------------------------------------
__global__ void k_intstr(const float* __restrict__ g, const float* __restrict__ th,
                         const float* __restrict__ cd, float* __restrict__ Iout) {
  __shared__ float red[NCELLS][256];
  const int tid = threadIdx.x;
  float rho[NCELLS];
#pragma unroll
  for (int c = 0; c < NCELLS; ++c) rho[c] = th[2 + c];
  float s[NCELLS] = {0.f, 0.f, 0.f, 0.f, 0.f};
  for (int n = tid; n < NN; n += 256) {
    float num[NCELLS], denom = 0.f;
#pragma unroll
    for (int c = 0; c < NCELLS; ++c) {
      num[c] = cd[n * NCELLS + c] * rho[c];
      denom += num[c];
    }
    const float w = g[n] / denom;
#pragma unroll
    for (int c = 0; c < NCELLS; ++c) s[c] += num[c] * w;
  }
#pragma unroll
  for (int c = 0; c < NCELLS; ++c) red[c][tid] = s[c];
  __syncthreads();
  for (int st = 128; st > 0; st >>= 1) {
    if (tid < st)
#pragma unroll
      for (int c = 0; c < NCELLS; ++c) red[c][tid] += red[c][tid + st];
    __syncthreads();
  }
  if (tid < NCELLS * NCELLS) {
    const int r = tid / NCELLS, c = tid % NCELLS;
    Iout[tid] = th[7] * th[8 + r] * red[c][0] / (float)NN;
  }
}

// ---------------------------------------------------------------------------
// 10) loss = 1 - cov(m,d) / (std(m)*std(d)) over the 75 concatenated values.
// ---------------------------------------------------------------------------
__global__ void k_loss(const float* __restrict__ Im, const float* __restrict__ Id,
                       float* __restrict__ out) {
  if (threadIdx.x == 0 && blockIdx.x == 0) {
    float sm = 0.f, sd = 0.f, smd = 0.f, sm2 = 0.f, sd2 = 0.f;
    for (int i = 0; i < 75; ++i) {
      const float m = Im[i], d = Id[i];
      sm += m; sd += d; smd += m * d; sm2 += m * m; sd2 += d * d;
    }
    const float inv = 1.0f / 75.0f;
    const float mm = sm * inv, md = sd * inv;
    const float cov  = smd * inv - mm * md;
    const float stdm = sqrtf(sm2 * inv - mm * mm);
    const float stdd = sqrtf(sd2 * inv - md * md);
    out[0] = 1.0f - cov / (stdm * stdd);
  }
}

// ---------------------------------------------------------------------------
// Orchestration
// ---------------------------------------------------------------------------
extern "C" void kernel_launch(void* const* d_in, const int* in_sizes, int n_in,
                              void* d_out, int out_size, void* d_ws, size_t ws_size,
                              hipStream_t stream) {
  (void)in_sizes; (void)n_in; (void)out_size; (void)ws_size;
  const float* th[3] = {(const float*)d_in[0], (const float*)d_in[1],
                        (const float*)d_in[2]};
  const float* cd   = (const float*)d_in[3];
  const float* Iexp = (const float*)d_in[4];
  const int*   nb   = (const int*)d_in[5];

  float* A   = (float*)d_ws;              // 4096*4096 f32 = 64 MB (fits in L2)
  float* vec = A + (size_t)NN * NN;       // b -> y -> g, in place
  float* Im  = vec + NN;                  // 3 x 25 model intensities

  for (int tix = 0; tix < 3; ++tix) {
    k_zero  <<<(NN * NN / 4) / 256, 256, 0, stream>>>((float4*)A);
    k_buildA<<<(NN + 255) / 256, 256, 0, stream>>>(A, th[tix], cd, nb);
    k_buildB<<<(NN + 255) / 256, 256, 0, stream>>>(vec, th[tix], cd);

    for (int K0 = 0; K0 < NN; K0 += NB) {
      k_panel<<<1, 512, 0, stream>>>(A, K0);
      const int M = NN - K0 - NB;
      if (M > 0) {
        k_trsm<<<(M + 255) / 256, 256, 0, stream>>>(A, K0);
        const int tiles_x = (M + 63) / 64;
        const int tiles_y = (M + 127) / 128;
        k_gemm<<<tiles_x * tiles_y, 256, 0, stream>>>(A, K0);
      }
    }

    k_fwd   <<<1, 256, 0, stream>>>(A, vec);
    k_bwd   <<<1, 256, 0, stream>>>(A, vec);
    k_intstr<<<1, 256, 0, stream>>>(vec, th[tix], cd, Im + 25 * tix);
  }
  k_loss<<<1, 32, 0, stream>>>(Im, Iexp, (float*)d_out);
}